// ScaledAttentionLayer_47193100648751
// MI455X (gfx1250) — compile-verified
//
#include <hip/hip_runtime.h>
#include <hip/hip_bf16.h>

// ScaledAttentionLayer fused kernel for MI455X (gfx1250, wave32).
//
//   scores[b,s] = (inputs[b,s,:] . weight) / sqrt(D)       (masked -> -2^31)
//   probs       = softmax_s(scores)
//   out[b,d]    = sum_s probs[b,s] * inputs[b,s,d]
//
// Bandwidth-bound (inputs = 839 MB >> 192 MB L2). Online softmax fuses both
// passes so each input element is read from HBM exactly once (~36 us floor at
// 23.3 TB/s). Pooling accumulate runs on V_WMMA_F32_16X16X4_F32.

typedef __attribute__((ext_vector_type(2))) float v2f;
typedef __attribute__((ext_vector_type(4))) float v4f;
typedef __attribute__((ext_vector_type(8))) float v8f;

#define BATCH 2048
#define SEQ   200
#define DIM   512
#define NEG_FILL (-2147483648.0f)   // -2^31, matches reference fill
#define INV_SQRT_D 0.04419417382415922f  // 1/sqrt(512)

__global__ __launch_bounds__(256) void
scaled_attn_pool_kernel(const float* __restrict__ x,          // [B,S,D]
                        const unsigned char* __restrict__ mask, // [B,S] bool
                        const float* __restrict__ weight,      // [D,1]
                        float* __restrict__ out)               // [B,D]
{
    __shared__ __align__(16) float lds_x[8][DIM]; // 8 staged rows (16 KB)
    __shared__ float lds_s[8];                    // 8 masked scaled scores

    const int b    = blockIdx.x;
    const int tid  = threadIdx.x;
    const int wave = tid >> 5;   // 0..7, one row of each 8-row chunk
    const int lane = tid & 31;

    // Per-lane weight fragment: lane holds w[(i*32+lane)*4 .. +3], i=0..3
    v4f wfrag[4];
    const v4f* w4 = (const v4f*)weight;
#pragma unroll
    for (int i = 0; i < 4; ++i)
        wfrag[i] = w4[i * 32 + lane];

    const float* xb = x + (size_t)b * SEQ * DIM;

    // Online-softmax running state (identical in every thread of the block).
    float m_run = NEG_FILL;
    float l_run = 0.0f;

    // Each wave owns 4 N-tiles of 16 columns: columns [wave*64, wave*64+64).
    v8f c[4];
#pragma unroll
    for (int t = 0; t < 4; ++t)
#pragma unroll
        for (int e = 0; e < 8; ++e)
            c[t][e] = 0.0f;

    const int nbase_wave = wave * 64;

    for (int chunk = 0; chunk < SEQ / 8; ++chunk) {
        const int srow = chunk * 8 + wave;   // SEQ = 200 = 8*25 exactly

        // ---- Stream one row from HBM (non-temporal b128), stage to LDS,
        // ---- and accumulate the score dot-product partial per lane.
        const v4f* xr4 = (const v4f*)(xb + (size_t)srow * DIM);
        float part = 0.0f;
#pragma unroll
        for (int i = 0; i < 4; ++i) {
            v4f v = __builtin_nontemporal_load(&xr4[i * 32 + lane]);
            ((v4f*)&lds_x[wave][0])[i * 32 + lane] = v;
            part += v.x * wfrag[i].x + v.y * wfrag[i].y
                  + v.z * wfrag[i].z + v.w * wfrag[i].w;
        }
        // wave32 butterfly reduction
#pragma unroll
        for (int off = 16; off >= 1; off >>= 1)
            part += __shfl_xor(part, off, 32);
        if (lane == 0) {
            float sc = part * INV_SQRT_D;
            lds_s[wave] = mask[(size_t)b * SEQ + srow] ? sc : NEG_FILL;
        }
        __syncthreads();

        // ---- Online softmax update (redundant per thread, identical values)
        float p[8];
        float cmax = NEG_FILL;
#pragma unroll
        for (int j = 0; j < 8; ++j) {
            p[j] = lds_s[j];
            cmax = fmaxf(cmax, p[j]);
        }
        const float m_new = fmaxf(m_run, cmax);
        const float scale = __expf(m_run - m_new);
        float lsum = 0.0f;
#pragma unroll
        for (int j = 0; j < 8; ++j) {
            p[j] = __expf(p[j] - m_new);
            lsum += p[j];
        }
        l_run = l_run * scale + lsum;
        m_run = m_new;

        // rescale accumulators by exp(m_old - m_new)
#pragma unroll
        for (int t = 0; t < 4; ++t)
#pragma unroll
            for (int e = 0; e < 8; ++e)
                c[t][e] *= scale;

        // ---- WMMA pooling accumulate: out_row += p[4k..4k+3] @ X[4 x 16]
        // A (16x4 f32, v2f):  VGPR0: lanes0-15 K=0, lanes16-31 K=2
        //                     VGPR1: lanes0-15 K=1, lanes16-31 K=3
        // Only M=0 (lane 0 / lane 16) is nonzero -> C rows 1..15 stay 0.
#pragma unroll
        for (int ks = 0; ks < 2; ++ks) {
            v2f a;
            a.x = (lane == 0) ? p[4 * ks + 0]
                : ((lane == 16) ? p[4 * ks + 2] : 0.0f);
            a.y = (lane == 0) ? p[4 * ks + 1]
                : ((lane == 16) ? p[4 * ks + 3] : 0.0f);
            // B (4x16 f32, v2f): VGPR0: lanes0-15 row K, lanes16-31 row K+2
            //                    VGPR1: rows K+1 / K+3
            const int krow = 4 * ks + ((lane < 16) ? 0 : 2);
            const int ncol = lane & 15;
#pragma unroll
            for (int t = 0; t < 4; ++t) {
                v2f bm;
                bm.x = lds_x[krow    ][nbase_wave + t * 16 + ncol];
                bm.y = lds_x[krow + 1][nbase_wave + t * 16 + ncol];
                c[t] = __builtin_amdgcn_wmma_f32_16x16x4_f32(
                    /*neg_a=*/false, a, /*neg_b=*/false, bm,
                    /*c_mod=*/(short)0, c[t],
                    /*reuse_a=*/false, /*reuse_b=*/false);
            }
        }
        __syncthreads();   // protect LDS rows before next chunk's stores
    }

    // ---- Write out: row M=0 of each C tile = element 0, lanes 0..15
    const float inv_l = 1.0f / l_run;
    if (lane < 16) {
#pragma unroll
        for (int t = 0; t < 4; ++t)
            out[(size_t)b * DIM + nbase_wave + t * 16 + lane] = c[t][0] * inv_l;
    }
}

extern "C" void kernel_launch(void* const* d_in, const int* in_sizes, int n_in,
                              void* d_out, int out_size, void* d_ws, size_t ws_size,
                              hipStream_t stream) {
    const float*         x    = (const float*)d_in[0];          // [B,S,D] fp32
    const unsigned char* mask = (const unsigned char*)d_in[1];  // [B,S] bool (1B)
    const float*         w    = (const float*)d_in[2];          // [D,1] fp32
    float*               out  = (float*)d_out;                  // [B,D] fp32

    dim3 grid(BATCH);
    dim3 block(256);   // 8 wave32 waves
    hipLaunchKernelGGL(scaled_attn_pool_kernel, grid, block, 0, stream,
                       x, mask, w, out);
}